// IouLoss_19318762898076
// MI455X (gfx1250) — compile-verified
//
#include <hip/hip_runtime.h>
#include <math.h>

// -------------------------------------------------------------------------
// IoU loss (CenterPoint-style aligned 3D IoU -> L1 on iou_pred).
// B=16, C=1, H=W=512, M=500, DIM=7.  8000 independent box pairs.
// Latency-bound gather + scalar geometry; WMMA f32 16x16x4 used as the
// wave32 cross-lane reduction engine (B = ones -> row sums).
// -------------------------------------------------------------------------

typedef __attribute__((ext_vector_type(2))) float v2f;
typedef __attribute__((ext_vector_type(8))) float v8f;

#define HW_CONST   (512 * 512)
#define M_CONST    500
#define D_CONST    7
#define NBLOCKS    32
#define BLOCK      256

// Wave32 all-lane sum via V_WMMA_F32_16X16X4_F32.
// A: lane's value in VGPR0, 0 in VGPR1 -> each lane value appears exactly
// once in the 16x4 A matrix. B = all ones (layout-independent).
// D[m][n] = rowsum_m; lanes 0-15 hold rows 0-7, lanes 16-31 rows 8-15,
// so per-lane sum of the 8 D registers + shfl_xor(16) = full 32-lane sum.
__device__ __forceinline__ float wave_sum_wmma(float v) {
    v2f a;  a.x = v;    a.y = 0.0f;
    v2f b;  b.x = 1.0f; b.y = 1.0f;
    v8f c = {};
    v8f d = __builtin_amdgcn_wmma_f32_16x16x4_f32(
        /*neg_a=*/false, a, /*neg_b=*/false, b,
        /*c_mod=*/(short)0, c, /*reuse_a=*/false, /*reuse_b=*/false);
    float t = ((d[0] + d[1]) + (d[2] + d[3])) + ((d[4] + d[5]) + (d[6] + d[7]));
    t += __shfl_xor(t, 16, 32);
    return t;
}

// BEV corners of box (x,y,z,dx,dy,dz,heading); CCW order.
__device__ __forceinline__ void bev_corners(const float* b, float* cx, float* cy) {
    const float hx[4] = { 0.5f, -0.5f, -0.5f,  0.5f };
    const float hy[4] = { 0.5f,  0.5f, -0.5f, -0.5f };
    float co = cosf(b[6]);
    float si = sinf(b[6]);
#pragma unroll
    for (int i = 0; i < 4; ++i) {
        float lx = hx[i] * b[3];
        float ly = hy[i] * b[4];
        cx[i] = lx * co - ly * si + b[0];
        cy[i] = lx * si + ly * co + b[1];
    }
}

// Intersection area of two convex quads via Sutherland-Hodgman clipping.
__device__ float quad_intersect_area(const float* ax, const float* ay,
                                     const float* bx, const float* by) {
    float px[12], py[12], qx[12], qy[12];
    int n = 4;
#pragma unroll
    for (int i = 0; i < 4; ++i) { px[i] = ax[i]; py[i] = ay[i]; }

#pragma unroll
    for (int e = 0; e < 4; ++e) {
        float x1 = bx[e],           y1 = by[e];
        float x2 = bx[(e + 1) & 3], y2 = by[(e + 1) & 3];
        float ex = x2 - x1, ey = y2 - y1;
        int m = 0;
        for (int i = 0; i < n; ++i) {
            int j = (i + 1 == n) ? 0 : i + 1;
            // signed distance to CCW edge: >=0 is inside
            float di = ex * (py[i] - y1) - ey * (px[i] - x1);
            float dj = ex * (py[j] - y1) - ey * (px[j] - x1);
            if (di >= 0.0f) { qx[m] = px[i]; qy[m] = py[i]; ++m; }
            if ((di >= 0.0f) != (dj >= 0.0f)) {
                float t = di / (di - dj);
                qx[m] = px[i] + t * (px[j] - px[i]);
                qy[m] = py[i] + t * (py[j] - py[i]);
                ++m;
            }
        }
        n = m;
        for (int i = 0; i < n; ++i) { px[i] = qx[i]; py[i] = qy[i]; }
        if (n == 0) break;
    }
    if (n < 3) return 0.0f;
    float area = 0.0f;
    for (int i = 0; i < n; ++i) {
        int j = (i + 1 == n) ? 0 : i + 1;
        area += px[i] * py[j] - px[j] * py[i];
    }
    return 0.5f * fabsf(area);
}

__device__ float aligned_iou3d(const float* a, const float* b) {
    float ax[4], ay[4], bx[4], by[4];
    bev_corners(a, ax, ay);
    bev_corners(b, bx, by);
    float inter_bev = quad_intersect_area(ax, ay, bx, by);
    float top = fminf(a[2] + a[5] * 0.5f, b[2] + b[5] * 0.5f);
    float bot = fmaxf(a[2] - a[5] * 0.5f, b[2] - b[5] * 0.5f);
    float inter = inter_bev * fmaxf(top - bot, 0.0f);
    float va = a[3] * a[4] * a[5];
    float vb = b[3] * b[4] * b[5];
    return inter / fmaxf(va + vb - inter, 1e-6f);
}

__global__ __launch_bounds__(BLOCK)
void iou_loss_main(const float* __restrict__ iou_pred,
                   const int*   __restrict__ mask,
                   const int*   __restrict__ ind,
                   const float* __restrict__ box_pred,
                   const float* __restrict__ box_gt,
                   float*       __restrict__ partials,
                   int N) {
    int t = blockIdx.x * BLOCK + threadIdx.x;

    float num = 0.0f, den = 0.0f;
    if (t < N) {
        int b   = t / M_CONST;
        int idx = ind[t];
        float w = (float)mask[t];

        // _transpose_and_gather_feat, C=1
        float pred = iou_pred[(size_t)b * HW_CONST + idx];

        // gather 7 channels of box_pred (BCHW): independent strided loads,
        // issued back-to-back so LOADcnt overlaps their latency.
        float pb[D_CONST];
        size_t base = (size_t)b * D_CONST * HW_CONST + (size_t)idx;
#pragma unroll
        for (int c = 0; c < D_CONST; ++c)
            pb[c] = box_pred[base + (size_t)c * HW_CONST];

        float gb[D_CONST];
        const float* g = box_gt + (size_t)t * D_CONST;
#pragma unroll
        for (int c = 0; c < D_CONST; ++c) gb[c] = g[c];

        float iou    = aligned_iou3d(pb, gb);
        float target = 2.0f * iou - 1.0f;
        num = fabsf(pred - target) * w;
        den = w;
    }

    // All lanes reconverged here: EXEC all-1s -> WMMA legal.
    float sn = wave_sum_wmma(num);
    float sd = wave_sum_wmma(den);

    __shared__ float sN[BLOCK / 32];
    __shared__ float sD[BLOCK / 32];
    int lane = threadIdx.x & 31;
    int wave = threadIdx.x >> 5;
    if (lane == 0) { sN[wave] = sn; sD[wave] = sd; }
    __syncthreads();

    if (threadIdx.x == 0) {
        float an = 0.0f, ad = 0.0f;
#pragma unroll
        for (int i = 0; i < BLOCK / 32; ++i) { an += sN[i]; ad += sD[i]; }
        partials[2 * blockIdx.x + 0] = an;   // fixed slot per block: deterministic
        partials[2 * blockIdx.x + 1] = ad;
    }
}

__global__ void iou_loss_finalize(const float* __restrict__ partials,
                                  float* __restrict__ out, int nblocks) {
    if (blockIdx.x == 0 && threadIdx.x == 0) {
        float n = 0.0f, d = 0.0f;
        for (int i = 0; i < nblocks; ++i) {
            n += partials[2 * i + 0];
            d += partials[2 * i + 1];
        }
        out[0] = n / (d + 1e-4f);
    }
}

extern "C" void kernel_launch(void* const* d_in, const int* in_sizes, int n_in,
                              void* d_out, int out_size, void* d_ws, size_t ws_size,
                              hipStream_t stream) {
    const float* iou_pred = (const float*)d_in[0];
    const int*   mask     = (const int*)  d_in[1];
    const int*   ind      = (const int*)  d_in[2];
    const float* box_pred = (const float*)d_in[3];
    const float* box_gt   = (const float*)d_in[4];
    float*       out      = (float*)d_out;

    int N = in_sizes[1];                 // B*M = 8000
    float* partials = (float*)d_ws;      // 2 floats per block, fully overwritten

    iou_loss_main<<<NBLOCKS, BLOCK, 0, stream>>>(
        iou_pred, mask, ind, box_pred, box_gt, partials, N);
    iou_loss_finalize<<<1, 32, 0, stream>>>(partials, out, NBLOCKS);
}